// Verse_25941602468492
// MI455X (gfx1250) — compile-verified
//
#include <hip/hip_runtime.h>
#include <cmath>

// Verse embedding SGD step for MI455X (gfx1250, wave32).
// Dot products batched 16-per-wave through V_WMMA_F32_16X16X4_F32
// (diagonal of the 16x16 accumulator = the 16 pair dots).
// Scatter-adds via global_atomic_add_f32; JAX snapshot semantics preserved
// by reading update operands from a frozen copy of the table in d_ws.

#define DIMS 128
#define LRATE 0.025f
#define LUT_N 1202

typedef float v2f __attribute__((ext_vector_type(2)));
typedef float v8f __attribute__((ext_vector_type(8)));

__device__ __forceinline__ float bcast_lane(float v, int srcLane) {
    // wave32 broadcast via DS_BPERMUTE_B32 (byte index, bits [6:2] used)
    return __int_as_float(
        __builtin_amdgcn_ds_bpermute(srcLane << 2, __float_as_int(v)));
}

// Grid-stride float4 copy: src -> dst0 (and dst1 if non-null).
__global__ void __launch_bounds__(256)
verse_copy_kernel(const float4* __restrict__ src, float4* __restrict__ dst0,
                  float4* __restrict__ dst1, long n4) {
    long i = (long)blockIdx.x * blockDim.x + threadIdx.x;
    long stride = (long)gridDim.x * blockDim.x;
    for (; i < n4; i += stride) {
        float4 v = src[i];
        dst0[i] = v;
        if (dst1) dst1[i] = v;
    }
}

// One wave handles 16 (u,v) pairs.
//   Wsnap : frozen table (reads)          Wout : live table (atomic adds)
//   isPos : 1 -> g = (1-sig)*LR ; 0 -> g = -sig*LR
__global__ void __launch_bounds__(256)
verse_pairs_kernel(const float* __restrict__ Wsnap,
                   float* __restrict__ Wout,
                   const float* __restrict__ lut,
                   const int* __restrict__ idxU,
                   const int* __restrict__ idxV,
                   int nPairs, float bias, int isPos) {
    const int lane = threadIdx.x & 31;
    const int wave = threadIdx.x >> 5;
    const int base = (blockIdx.x * 8 + wave) * 16;   // 16 pairs per wave
    if (base >= nPairs) return;                       // whole-wave exit, EXEC stays full

    // Fragment mapping for V_WMMA_F32_16X16X4_F32:
    //   lane = p + 16h ; this lane supplies row elements [4c+2h, 4c+2h+1]
    //   for both the A (u-rows, 16x4) and B (v-rows^T, 4x16) fragments.
    const int p = lane & 15;
    const int h = lane >> 4;
    int pg = base + p;
    if (pg >= nPairs) pg = nPairs - 1;               // clamp: WMMA needs EXEC all-1s
    const int u = idxU[pg];
    const int v = idxV[pg];
    const float* uptr = Wsnap + (size_t)u * DIMS + 2 * h;
    const float* vptr = Wsnap + (size_t)v * DIMS + 2 * h;

    v8f acc0 = {0.f, 0.f, 0.f, 0.f, 0.f, 0.f, 0.f, 0.f};
    v8f acc1 = {0.f, 0.f, 0.f, 0.f, 0.f, 0.f, 0.f, 0.f};
#pragma unroll
    for (int c = 0; c < 32; c += 2) {                // K = 128 in chunks of 4
        v2f a0 = *(const v2f*)(uptr + 4 * c);
        v2f b0 = *(const v2f*)(vptr + 4 * c);
        acc0 = __builtin_amdgcn_wmma_f32_16x16x4_f32(false, a0, false, b0,
                                                     (short)0, acc0, false, false);
        v2f a1 = *(const v2f*)(uptr + 4 * (c + 1));
        v2f b1 = *(const v2f*)(vptr + 4 * (c + 1));
        acc1 = __builtin_amdgcn_wmma_f32_16x16x4_f32(false, a1, false, b1,
                                                     (short)0, acc1, false, false);
    }
    v8f acc = acc0 + acc1;                           // C[i][j] = dot(U_i, V_j)

    // Diagonal C[p][p]: VGPR (p&7), lane p (p<8) or lane p+16 (p>=8).
#pragma unroll
    for (int q = 0; q < 16; ++q) {
        const int srcLane = (q < 8) ? q : (q + 16);
        float dot = bcast_lane(acc[q & 7], srcLane); // broadcast to all 32 lanes

        float s = dot - bias;
        s = fminf(6.0f, fmaxf(-6.0f, s));
        int li = (int)floorf((s + 6.01f) / 0.01f);
        li = li < 0 ? 0 : (li > LUT_N - 1 ? LUT_N - 1 : li);
        float sig = lut[li];
        float g = isPos ? (1.0f - sig) * LRATE : (0.0f - sig * LRATE);

        const int pid = base + q;
        if (pid < nPairs) {                          // tail mask (no WMMA here)
            const int uu = idxU[pid];
            const int vv = idxV[pid];
            // 32 lanes x 4 elems = 128-dim row; operands from the snapshot.
            const float4 ou = *(const float4*)(Wsnap + (size_t)uu * DIMS + lane * 4);
            const float4 ov = *(const float4*)(Wsnap + (size_t)vv * DIMS + lane * 4);
            float* du = Wout + (size_t)uu * DIMS + lane * 4;
            float* dv = Wout + (size_t)vv * DIMS + lane * 4;
            atomicAdd(du + 0, g * ov.x);
            atomicAdd(du + 1, g * ov.y);
            atomicAdd(du + 2, g * ov.z);
            atomicAdd(du + 3, g * ov.w);
            atomicAdd(dv + 0, g * ou.x);
            atomicAdd(dv + 1, g * ou.y);
            atomicAdd(dv + 2, g * ou.z);
            atomicAdd(dv + 3, g * ou.w);
        }
    }
}

extern "C" void kernel_launch(void* const* d_in, const int* in_sizes, int n_in,
                              void* d_out, int out_size, void* d_ws, size_t ws_size,
                              hipStream_t stream) {
    const float* W   = (const float*)d_in[0];
    const float* lut = (const float*)d_in[1];
    const int* ipu   = (const int*)d_in[2];
    const int* ipv   = (const int*)d_in[3];
    const int* inu   = (const int*)d_in[4];
    const int* inv   = (const int*)d_in[5];
    float* Wout  = (float*)d_out;
    float* Wsnap = (float*)d_ws;                     // 51.2 MB snapshot

    const int nPos = in_sizes[2];                    // N
    const int nNeg = in_sizes[4];                    // N*NEG
    const long nElem = (long)in_sizes[0];            // N*D
    const long n4 = nElem / 4;
    const int nNodes = (int)(nElem / DIMS);
    const int negPer = nNeg / (nPos > 0 ? nPos : 1);
    const float nceBias    = logf((float)nNodes);
    const float nceNegBias = logf((float)nNodes / (float)(negPer > 0 ? negPer : 1));

    // Phase 0: W -> live table AND frozen snapshot.
    verse_copy_kernel<<<1024, 256, 0, stream>>>(
        (const float4*)W, (float4*)Wout, (float4*)Wsnap, n4);

    // Phase 1: positive pairs (reads snapshot, atomically updates live).
    const int blocksP = ((nPos + 15) / 16 + 7) / 8;
    verse_pairs_kernel<<<blocksP, 256, 0, stream>>>(
        Wsnap, Wout, lut, ipu, ipv, nPos, nceBias, 1);

    // Phase 2: re-snapshot the post-positive table.
    verse_copy_kernel<<<1024, 256, 0, stream>>>(
        (const float4*)Wout, (float4*)Wsnap, (float4*)nullptr, n4);

    // Phase 3: negative pairs.
    const int blocksN = ((nNeg + 15) / 16 + 7) / 8;
    verse_pairs_kernel<<<blocksN, 256, 0, stream>>>(
        Wsnap, Wout, lut, inu, inv, nNeg, nceNegBias, 0);
}